// RelationalGCN_46626164965721
// MI455X (gfx1250) — compile-verified
//
#include <hip/hip_runtime.h>
#include <hip/hip_bf16.h>

typedef __attribute__((ext_vector_type(2))) float v2f;
typedef __attribute__((ext_vector_type(8))) float v8f;
typedef int v4i __attribute__((vector_size(16)));

#define N_NODES 50000
#define N_EDGES 800000
#define DIN 1024
#define DHID 256
#define DOUT 128

#if defined(__gfx1250__) && __has_builtin(__builtin_amdgcn_global_load_async_to_lds_b128)
#define HAVE_ASYNC_LDS 1
#else
#define HAVE_ASYNC_LDS 0
#endif

typedef v4i __attribute__((address_space(1))) gas_v4i;   // global int4
typedef v4i __attribute__((address_space(3))) las_v4i;   // LDS int4

// 16B global -> LDS copy; async (ASYNCcnt-tracked, no VGPR round trip) when available.
__device__ __forceinline__ void g2l_b128(const float* gsrc, float* ldst)
{
#if HAVE_ASYNC_LDS
    __builtin_amdgcn_global_load_async_to_lds_b128(
        (gas_v4i*)gsrc, (las_v4i*)ldst, 0, 0);
#else
    *(float4*)ldst = *(const float4*)gsrc;
#endif
}

__device__ __forceinline__ void wait_async_lds()
{
#if HAVE_ASYNC_LDS
#if __has_builtin(__builtin_amdgcn_s_wait_asynccnt)
    __builtin_amdgcn_s_wait_asynccnt(0);
#else
    asm volatile("s_wait_asynccnt 0x0" ::: "memory");
#endif
#endif
}

// ---------------------------------------------------------------------------
// C[M,N] = A[M,K] x B[K,N] (+ bias[N]), row-major fp32.
// Block: 256 threads = 8 waves; block tile 16(M) x 128(N); wave tile 16x16.
// K consumed in 64-wide chunks staged to LDS (double-buffered, async copies),
// V_WMMA_F32_16X16X4_F32 with two independent accumulator chains for ILP.
// Requires M%16==0 (grid.y), N%128==0 (grid.x), K%64==0. No divergence ->
// EXEC all-ones for WMMA. LDS strides 68/132: 16B-aligned rows, stride%64=4
// spreads the 16 active lanes across distinct banks.
// ---------------------------------------------------------------------------
__global__ __launch_bounds__(256) void gemm_wmma_f32(
    const float* __restrict__ A, const float* __restrict__ B,
    const float* __restrict__ bias, float* __restrict__ C,
    int K, int N)
{
    __shared__ __align__(16) float lA[2][16 * 68];
    __shared__ __align__(16) float lB[2][64 * 132];

    const int t       = threadIdx.x;
    const int lane    = t & 31;
    const int wave    = t >> 5;
    const int rowBase = blockIdx.y * 16;
    const int colBase = blockIdx.x * 128;

    const int m  = lane & 15;               // A-frag row (ISA 16x4 A layout)
    const int kh = lane >> 4;               // lane group selects K pair
    const int nl = wave * 16 + (lane & 15); // local N column

    // Per-thread staging coordinates (fixed across chunks)
    const int aIdx = t * 4;
    const int aR   = aIdx >> 6;             // /64
    const int aC   = aIdx & 63;
    const float* aG = A + (size_t)(rowBase + aR) * K + aC;

    v8f acc0 = {}, acc1 = {};

    const int nchunks = K >> 6;

    // --- prologue: stage chunk 0 into buffer 0 ---
    g2l_b128(aG, &lA[0][aR * 68 + aC]);
    #pragma unroll
    for (int i = 0; i < 8; ++i) {
        int q = (t + i * 256) * 4;
        int r = q >> 7;                     // /128
        int c = q & 127;
        g2l_b128(B + (size_t)r * N + colBase + c, &lB[0][r * 132 + c]);
    }
    wait_async_lds();
    __syncthreads();

    for (int ci = 0; ci < nchunks; ++ci) {
        const int cur = ci & 1;
        // --- prefetch chunk ci+1 into the other buffer (async) ---
        if (ci + 1 < nchunks) {
            const int kc = (ci + 1) << 6;
            g2l_b128(aG + kc, &lA[cur ^ 1][aR * 68 + aC]);
            #pragma unroll
            for (int i = 0; i < 8; ++i) {
                int q = (t + i * 256) * 4;
                int r = q >> 7;
                int c = q & 127;
                g2l_b128(B + (size_t)(kc + r) * N + colBase + c,
                         &lB[cur ^ 1][r * 132 + c]);
            }
        }
        // --- compute on current buffer: 16 WMMAs, 2 independent chains ---
        const float* cA = lA[cur];
        const float* cB = lB[cur];
        #pragma unroll
        for (int kk = 0; kk < 64; kk += 8) {
            v2f a0, b0, a1, b1;
            a0.x = cA[m * 68 + kk + 2 * kh + 0];
            a0.y = cA[m * 68 + kk + 2 * kh + 1];
            b0.x = cB[(kk + 2 * kh + 0) * 132 + nl];
            b0.y = cB[(kk + 2 * kh + 1) * 132 + nl];
            acc0 = __builtin_amdgcn_wmma_f32_16x16x4_f32(
                false, a0, false, b0, (short)0, acc0, false, false);
            a1.x = cA[m * 68 + kk + 4 + 2 * kh + 0];
            a1.y = cA[m * 68 + kk + 4 + 2 * kh + 1];
            b1.x = cB[(kk + 4 + 2 * kh + 0) * 132 + nl];
            b1.y = cB[(kk + 4 + 2 * kh + 1) * 132 + nl];
            acc1 = __builtin_amdgcn_wmma_f32_16x16x4_f32(
                false, a1, false, b1, (short)0, acc1, false, false);
        }
        wait_async_lds();
        __syncthreads();
    }

    // --- epilogue: D layout = VGPR v -> row (v + 8*kh), col = lane&15 ---
    const int nc = colBase + wave * 16 + (lane & 15);
    const float bv = bias ? bias[nc] : 0.0f;
    #pragma unroll
    for (int v = 0; v < 8; ++v) {
        int mr = rowBase + v + 8 * kh;
        C[(size_t)mr * N + nc] = acc0[v] + acc1[v] + bv;
    }
}

// ---------------------------------------------------------------------------
// Per-edge message + scatter-add, float4-vectorized:
//   agg[dst] += a[et,0]*y0[src] + a[et,1]*y1[src]
// One thread handles 4 feature dims; D/4 is a power of two -> pass shift.
// ---------------------------------------------------------------------------
__global__ __launch_bounds__(256) void edge_scatter4(
    const float4* __restrict__ y0, const float4* __restrict__ y1,
    const float* __restrict__ coef,            // [R,2]
    const int* __restrict__ src, const int* __restrict__ dst,
    const int* __restrict__ et,
    float* __restrict__ agg, long total4, int d4shift)
{
    long tid = (long)blockIdx.x * 256 + threadIdx.x;
    if (tid >= total4) return;
    int e = (int)(tid >> d4shift);
    int d = (int)(tid & ((1 << d4shift) - 1));
    int s = src[e], tn = dst[e], r = et[e];
    float c0 = coef[2 * r + 0];
    float c1 = coef[2 * r + 1];
    size_t D4 = (size_t)1 << d4shift;
    float4 v0 = y0[(size_t)s * D4 + d];
    float4 v1 = y1[(size_t)s * D4 + d];
    size_t base = ((size_t)tn * D4 + d) * 4;
    atomicAdd(&agg[base + 0], c0 * v0.x + c1 * v1.x);
    atomicAdd(&agg[base + 1], c0 * v0.y + c1 * v1.y);
    atomicAdd(&agg[base + 2], c0 * v0.z + c1 * v1.z);
    atomicAdd(&agg[base + 3], c0 * v0.w + c1 * v1.w);
}

__global__ __launch_bounds__(256) void relu_inplace4(float4* __restrict__ p, long n4)
{
    long tid = (long)blockIdx.x * 256 + threadIdx.x;
    if (tid >= n4) return;
    float4 v = p[tid];
    v.x = fmaxf(v.x, 0.0f); v.y = fmaxf(v.y, 0.0f);
    v.z = fmaxf(v.z, 0.0f); v.w = fmaxf(v.w, 0.0f);
    p[tid] = v;
}

// out[n, 0:1024] = x[n];  out[n, 1024:1152] = relu(agg2[n]); float4 lanes.
__global__ __launch_bounds__(256) void concat_out4(
    const float4* __restrict__ x4, const float4* __restrict__ agg2_4,
    float4* __restrict__ out4, long total4)
{
    long tid = (long)blockIdx.x * 256 + threadIdx.x;
    if (tid >= total4) return;
    const int W = (DIN + DOUT) / 4;            // 288
    int n = (int)(tid / W);
    int c = (int)(tid % W);
    float4 v;
    if (c < DIN / 4) {
        v = x4[(size_t)n * (DIN / 4) + c];
    } else {
        v = agg2_4[(size_t)n * (DOUT / 4) + (c - DIN / 4)];
        v.x = fmaxf(v.x, 0.0f); v.y = fmaxf(v.y, 0.0f);
        v.z = fmaxf(v.z, 0.0f); v.w = fmaxf(v.w, 0.0f);
    }
    out4[tid] = v;
}

extern "C" void kernel_launch(void* const* d_in, const int* in_sizes, int n_in,
                              void* d_out, int out_size, void* d_ws, size_t ws_size,
                              hipStream_t stream) {
    const float* x   = (const float*)d_in[0];
    const int*   src = (const int*)  d_in[1];
    const int*   dst = (const int*)  d_in[2];
    const int*   ety = (const int*)  d_in[3];
    const float* V1  = (const float*)d_in[4];   // [2,1024,256]
    const float* a1  = (const float*)d_in[5];   // [5,2]
    const float* Ws1 = (const float*)d_in[6];   // [1024,256]
    const float* b1  = (const float*)d_in[7];   // [256]
    const float* V2  = (const float*)d_in[8];   // [2,256,128]
    const float* a2  = (const float*)d_in[9];   // [5,2]
    const float* Ws2 = (const float*)d_in[10];  // [256,128]
    const float* b2  = (const float*)d_in[11];  // [128]
    float* out = (float*)d_out;

    // Workspace (floats): 50000*(3*256 + 3*128) = 57.6M floats = 230.4 MB
    float* ws   = (float*)d_ws;
    float* yb0  = ws;                                  // [N,256] basis-0 proj
    float* yb1  = yb0  + (size_t)N_NODES * DHID;       // [N,256] basis-1 proj
    float* agg1 = yb1  + (size_t)N_NODES * DHID;       // [N,256] self+bias -> +msgs -> relu = h
    float* zb0  = agg1 + (size_t)N_NODES * DHID;       // [N,128]
    float* zb1  = zb0  + (size_t)N_NODES * DOUT;       // [N,128]
    float* agg2 = zb1  + (size_t)N_NODES * DOUT;       // [N,128]

    dim3 blk(256);

    // ---------------- Layer 1 dense phase ----------------
    dim3 g1(DHID / 128, N_NODES / 16);                 // 2 x 3125
    gemm_wmma_f32<<<g1, blk, 0, stream>>>(x, V1,                      nullptr, yb0,  DIN, DHID);
    gemm_wmma_f32<<<g1, blk, 0, stream>>>(x, V1 + (size_t)DIN * DHID, nullptr, yb1,  DIN, DHID);
    gemm_wmma_f32<<<g1, blk, 0, stream>>>(x, Ws1,                     b1,      agg1, DIN, DHID);

    // ---------------- Layer 1 edge phase ----------------
    {
        long total4 = (long)N_EDGES * (DHID / 4);      // 51.2M
        edge_scatter4<<<dim3((unsigned)((total4 + 255) / 256)), blk, 0, stream>>>(
            (const float4*)yb0, (const float4*)yb1, a1, src, dst, ety,
            agg1, total4, 6 /* log2(256/4) */);
    }
    {
        long n4 = (long)N_NODES * DHID / 4;
        relu_inplace4<<<dim3((unsigned)((n4 + 255) / 256)), blk, 0, stream>>>(
            (float4*)agg1, n4);
    }

    // ---------------- Layer 2 dense phase (input h = agg1) ----------------
    dim3 g2(DOUT / 128, N_NODES / 16);                 // 1 x 3125
    gemm_wmma_f32<<<g2, blk, 0, stream>>>(agg1, V2,                       nullptr, zb0,  DHID, DOUT);
    gemm_wmma_f32<<<g2, blk, 0, stream>>>(agg1, V2 + (size_t)DHID * DOUT, nullptr, zb1,  DHID, DOUT);
    gemm_wmma_f32<<<g2, blk, 0, stream>>>(agg1, Ws2,                      b2,      agg2, DHID, DOUT);

    // ---------------- Layer 2 edge phase ----------------
    {
        long total4 = (long)N_EDGES * (DOUT / 4);      // 25.6M
        edge_scatter4<<<dim3((unsigned)((total4 + 255) / 256)), blk, 0, stream>>>(
            (const float4*)zb0, (const float4*)zb1, a2, src, dst, ety,
            agg2, total4, 5 /* log2(128/4) */);
    }

    // ---------------- Output: concat([x, relu(agg2)]) ----------------
    {
        long total4 = (long)N_NODES * (DIN + DOUT) / 4; // 14.4M
        concat_out4<<<dim3((unsigned)((total4 + 255) / 256)), blk, 0, stream>>>(
            (const float4*)x, (const float4*)agg2, (float4*)out, total4);
    }
}